// TransitionDownBlock_76381698392664
// MI455X (gfx1250) — compile-verified
//
#include <hip/hip_runtime.h>
#include <hip/hip_bf16.h>
#include <float.h>

typedef __attribute__((ext_vector_type(2))) float v2f;
typedef __attribute__((ext_vector_type(8))) float v8f;

#define B_    8
#define N_    8192
#define CIN_  64
#define COUT_ 128
#define S_    2048
#define K_    16

// ---------------------------------------------------------------------------
// GEMM (1x1 conv) + deterministic per-block BN partial stats.
// Wave computes a full 128(M) x 16(N) output tile via V_WMMA_F32_16X16X4_F32.
// A-frag layout (32-bit A 16x4): lane l, vgpr j -> A[m = l%16][k = 2*(l/16)+j]
// B-frag layout (assumed mirror): lane l, vgpr j -> B[k = 2*(l/16)+j][n = l%16]
// C/D layout: lane l, vgpr j -> D[m = 8*(l/16)+j][n = l%16]
// ---------------------------------------------------------------------------
template<int CIN, bool STAGE_W, bool TRANSFORM_IN>
__global__ __launch_bounds__(128)
void gemm_kernel(const float* __restrict__ X,        // [B, CIN, N]
                 const float* __restrict__ W,        // [128, CIN]
                 const float* __restrict__ bias,     // [128]
                 const float* __restrict__ scale_in, // [CIN] (BN fold) or null
                 const float* __restrict__ shift_in, // [CIN] or null
                 float* __restrict__ Y,              // [B, 128, N]
                 float* __restrict__ part_sum,       // [numBlocks][128]
                 float* __restrict__ part_sq)        // [numBlocks][128]
{
  constexpr int CINP = CIN + 2;                  // LDS row pad -> no bank conflicts
  constexpr int LWSZ = STAGE_W ? 128 * CINP : 1;
  __shared__ float lw[LWSZ];
  __shared__ float lsum[4 * 128];
  __shared__ float lsq [4 * 128];

  const int tid  = threadIdx.x;
  const int lane = tid & 31;
  const int wv   = tid >> 5;
  const int lm   = lane & 15;
  const int lh   = lane >> 4;
  const int b    = blockIdx.y;
  const int nb   = blockIdx.x * 64 + wv * 16;

  if (STAGE_W) {
    for (int idx = tid; idx < 128 * CIN; idx += 128) {
      int r = idx / CIN, c = idx - r * CIN;
      lw[r * CINP + c] = W[idx];
    }
    __syncthreads();
  }

  v8f acc[8] = {};

  const float* xb = X + (size_t)b * CIN * N_;

  for (int kk = 0; kk < CIN / 4; ++kk) {
    const int k0 = kk * 4;
    const int c0 = k0 + 2 * lh;
    // ---- B fragment (per-point activations), with optional fused BN1+ReLU
    float bv0 = xb[(size_t)c0 * N_ + nb + lm];
    float bv1 = xb[(size_t)(c0 + 1) * N_ + nb + lm];
    if (TRANSFORM_IN) {
      bv0 = fmaxf(fmaf(scale_in[c0],     bv0, shift_in[c0]),     0.0f);
      bv1 = fmaxf(fmaf(scale_in[c0 + 1], bv1, shift_in[c0 + 1]), 0.0f);
    }
    v2f bf; bf.x = bv0; bf.y = bv1;
    // ---- 8 M-tiles of A (weights) x B
#pragma unroll
    for (int mB = 0; mB < 8; ++mB) {
      v2f af;
      if (STAGE_W) af = *(const v2f*)&lw[(mB * 16 + lm) * CINP + c0];
      else         af = *(const v2f*)&W [(mB * 16 + lm) * CIN  + c0];
      acc[mB] = __builtin_amdgcn_wmma_f32_16x16x4_f32(
          false, af, false, bf, (short)0, acc[mB], false, false);
    }
  }

  // ---- bias, store, deterministic per-channel partial sums ----
#pragma unroll
  for (int mB = 0; mB < 8; ++mB) {
#pragma unroll
    for (int j = 0; j < 8; ++j) {
      const int m = mB * 16 + lh * 8 + j;
      const float y = acc[mB][j] + bias[m];
      Y[((size_t)b * 128 + m) * N_ + nb + lm] = y;
      float s = y, q = y * y;
#pragma unroll
      for (int msk = 1; msk < 16; msk <<= 1) {   // reduce over the 16 N-lanes
        s += __shfl_xor(s, msk);
        q += __shfl_xor(q, msk);
      }
      if (lm == 0) { lsum[wv * 128 + m] = s; lsq[wv * 128 + m] = q; }
    }
  }
  __syncthreads();
  if (tid < 128) {
    const float s = lsum[tid] + lsum[128 + tid] + lsum[256 + tid] + lsum[384 + tid];
    const float q = lsq [tid] + lsq [128 + tid] + lsq [256 + tid] + lsq [384 + tid];
    const int blk = blockIdx.y * gridDim.x + blockIdx.x;
    part_sum[(size_t)blk * 128 + tid] = s;
    part_sq [(size_t)blk * 128 + tid] = q;
  }
}

// ---------------------------------------------------------------------------
// Reduce block partials in fixed order (deterministic), produce BN scale/shift.
// ---------------------------------------------------------------------------
__global__ __launch_bounds__(128)
void finalize_kernel(const float* __restrict__ part_sum,
                     const float* __restrict__ part_sq, int nparts,
                     const float* __restrict__ gamma, const float* __restrict__ beta,
                     float* __restrict__ scale, float* __restrict__ shift, float count)
{
  const int c = threadIdx.x;
  float s = 0.f, q = 0.f;
  for (int i = 0; i < nparts; ++i) {
    s += part_sum[(size_t)i * 128 + c];
    q += part_sq [(size_t)i * 128 + c];
  }
  const float mean = s / count;
  const float var  = q / count - mean * mean;
  const float sc   = gamma[c] * rsqrtf(var + 1e-5f);
  scale[c] = sc;
  shift[c] = beta[c] - mean * sc;
}

// ---------------------------------------------------------------------------
// Farthest point sampling: one 1024-thread block per batch, 8 pts/thread in
// VGPRs, argmax via wave shuffle butterfly + LDS tree. Tie-break: lowest index
// (matches jnp.argmax). Writes fps indices + centroids ([B,3,S] in d_out tail).
// ---------------------------------------------------------------------------
__global__ __launch_bounds__(1024)
void fps_kernel(const float* __restrict__ pts,   // [B, 3, N]
                int* __restrict__ fps_idx,       // [B, S]
                float* __restrict__ cent)        // [B, 3, S]
{
  const int b = blockIdx.x;
  const int t = threadIdx.x;
  const float* px_ = pts + (size_t)b * 3 * N_;
  const float* py_ = px_ + N_;
  const float* pz_ = px_ + 2 * N_;

  float px[8], py[8], pz[8], dist[8];
#pragma unroll
  for (int j = 0; j < 8; ++j) {
    const int i = t + j * 1024;
    px[j] = px_[i]; py[j] = py_[i]; pz[j] = pz_[i];
  }
  const float s0x = px_[0], s0y = py_[0], s0z = pz_[0];
#pragma unroll
  for (int j = 0; j < 8; ++j) {
    const float dx = px[j] - s0x, dy = py[j] - s0y, dz = pz[j] - s0z;
    dist[j] = dx * dx + dy * dy + dz * dz;
  }
  if (t == 0) {
    fps_idx[b * S_] = 0;
    cent[(size_t)b * 3 * S_           ] = s0x;
    cent[(size_t)b * 3 * S_ +     S_  ] = s0y;
    cent[(size_t)b * 3 * S_ + 2 * S_  ] = s0z;
  }

  __shared__ float red_d[32];
  __shared__ int   red_i[32];
  __shared__ float selc[3];
  const int lane = t & 31, wv = t >> 5;

  for (int s = 1; s < S_; ++s) {
    float bd = dist[0]; int bi = t;
#pragma unroll
    for (int j = 1; j < 8; ++j)
      if (dist[j] > bd) { bd = dist[j]; bi = t + j * 1024; }
#pragma unroll
    for (int msk = 1; msk < 32; msk <<= 1) {
      const float od = __shfl_xor(bd, msk);
      const int   oi = __shfl_xor(bi, msk);
      if (od > bd || (od == bd && oi < bi)) { bd = od; bi = oi; }
    }
    if (lane == 0) { red_d[wv] = bd; red_i[wv] = bi; }
    __syncthreads();
    if (wv == 0) {
      bd = red_d[lane]; bi = red_i[lane];
#pragma unroll
      for (int msk = 1; msk < 32; msk <<= 1) {
        const float od = __shfl_xor(bd, msk);
        const int   oi = __shfl_xor(bi, msk);
        if (od > bd || (od == bd && oi < bi)) { bd = od; bi = oi; }
      }
      if (lane == 0) {
        const float cx = px_[bi], cy = py_[bi], cz = pz_[bi];
        selc[0] = cx; selc[1] = cy; selc[2] = cz;
        fps_idx[b * S_ + s] = bi;
        cent[(size_t)b * 3 * S_          + s] = cx;
        cent[(size_t)b * 3 * S_ +     S_ + s] = cy;
        cent[(size_t)b * 3 * S_ + 2 * S_ + s] = cz;
      }
    }
    __syncthreads();
    const float cx = selc[0], cy = selc[1], cz = selc[2];
#pragma unroll
    for (int j = 0; j < 8; ++j) {
      const float dx = px[j] - cx, dy = py[j] - cy, dz = pz[j] - cz;
      const float d = dx * dx + dy * dy + dz * dz;
      dist[j] = fminf(dist[j], d);
    }
  }
}

// ---------------------------------------------------------------------------
// kNN top-16: one wave per (b,s). Each lane keeps a sorted top-16 in registers
// (unrolled insertion), dumps to LDS, then a 32-way merge with wave-wide
// min-pair shuffles. (dist, idx) lexicographic order => matches stable top_k.
// ---------------------------------------------------------------------------
__global__ __launch_bounds__(256)
void knn_kernel(const float* __restrict__ pts,   // [B, 3, N]
                const float* __restrict__ cent,  // [B, 3, S]
                int* __restrict__ knn_idx)       // [B*S, K]
{
  __shared__ float kd[8 * 512];
  __shared__ int   ki[8 * 512];
  const int lane = threadIdx.x & 31;
  const int wv   = threadIdx.x >> 5;
  const int item = blockIdx.x * 8 + wv;     // (b*S + s)
  const int b = item / S_;
  const int s = item - b * S_;

  const float* px_ = pts + (size_t)b * 3 * N_;
  const float cx = cent[(size_t)b * 3 * S_          + s];
  const float cy = cent[(size_t)b * 3 * S_ +     S_ + s];
  const float cz = cent[(size_t)b * 3 * S_ + 2 * S_ + s];
  const float c2 = cx * cx + cy * cy + cz * cz;

  float ld[16]; int li[16];
#pragma unroll
  for (int q = 0; q < 16; ++q) { ld[q] = FLT_MAX; li[q] = 0x7fffffff; }

  for (int i = lane; i < N_; i += 32) {
    const float x = px_[i], y = px_[N_ + i], z = px_[2 * N_ + i];
    const float p2 = x * x + y * y + z * z;
    const float d  = c2 + p2 - 2.0f * (cx * x + cy * y + cz * z);
    if (d < ld[15] || (d == ld[15] && i < li[15])) {
      float cd = d; int ci = i;
#pragma unroll
      for (int q = 0; q < 16; ++q) {
        const bool sw = (cd < ld[q]) || (cd == ld[q] && ci < li[q]);
        const float nd = sw ? ld[q] : cd;
        const int   ni = sw ? li[q] : ci;
        if (sw) { ld[q] = cd; li[q] = ci; }
        cd = nd; ci = ni;
      }
    }
  }

  const int base = wv * 512 + lane * 16;
#pragma unroll
  for (int q = 0; q < 16; ++q) { kd[base + q] = ld[q]; ki[base + q] = li[q]; }
  __syncthreads();

  int head = 0;
  int* out = knn_idx + (size_t)item * K_;
  for (int j = 0; j < K_; ++j) {
    float hd; int hi;
    if (head < 16) { hd = kd[base + head]; hi = ki[base + head]; }
    else           { hd = FLT_MAX;         hi = 0x7fffffff;      }
    float bd = hd; int bi = hi; int bl = lane;
#pragma unroll
    for (int msk = 1; msk < 32; msk <<= 1) {
      const float od = __shfl_xor(bd, msk);
      const int   oi = __shfl_xor(bi, msk);
      const int   ol = __shfl_xor(bl, msk);
      if (od < bd || (od == bd && oi < bi)) { bd = od; bi = oi; bl = ol; }
    }
    if (lane == 0) out[j] = bi;
    if (lane == bl) ++head;
  }
}

// ---------------------------------------------------------------------------
// Gather neighbors, apply fused BN2+ReLU, mean over K. block = 128 channels.
// ---------------------------------------------------------------------------
__global__ __launch_bounds__(128)
void gather_kernel(const float* __restrict__ y2,      // [B, 128, N]
                   const float* __restrict__ scale2,
                   const float* __restrict__ shift2,
                   const int* __restrict__ knn_idx,   // [B*S, K]
                   float* __restrict__ out)           // [B, 128, S]
{
  const int s = blockIdx.x, b = blockIdx.y, c = threadIdx.x;
  const float sc = scale2[c], sh = shift2[c];
  const float* yb = y2 + ((size_t)b * 128 + c) * N_;
  const int* nbi = knn_idx + (size_t)(b * S_ + s) * K_;
  float acc = 0.f;
#pragma unroll
  for (int k = 0; k < K_; ++k) {
    const float v = yb[nbi[k]];
    acc += fmaxf(fmaf(sc, v, sh), 0.0f);
  }
  out[((size_t)b * 128 + c) * S_ + s] = acc * (1.0f / K_);
}

// ---------------------------------------------------------------------------
extern "C" void kernel_launch(void* const* d_in, const int* in_sizes, int n_in,
                              void* d_out, int out_size, void* d_ws, size_t ws_size,
                              hipStream_t stream)
{
  const float* feats  = (const float*)d_in[0];
  const float* points = (const float*)d_in[1];
  const float* W1 = (const float*)d_in[2];
  const float* b1 = (const float*)d_in[3];
  const float* g1 = (const float*)d_in[4];
  const float* be1 = (const float*)d_in[5];
  const float* W2 = (const float*)d_in[6];
  const float* b2 = (const float*)d_in[7];
  const float* g2 = (const float*)d_in[8];
  const float* be2 = (const float*)d_in[9];

  float* out  = (float*)d_out;
  float* cent = out + (size_t)B_ * COUT_ * S_;   // centroids appended after out

  float* ws = (float*)d_ws;
  size_t off = 0;
  float* y1    = ws + off; off += (size_t)B_ * COUT_ * N_;
  float* y2    = ws + off; off += (size_t)B_ * COUT_ * N_;
  float* psum1 = ws + off; off += 1024 * 128;
  float* psq1  = ws + off; off += 1024 * 128;
  float* psum2 = ws + off; off += 1024 * 128;
  float* psq2  = ws + off; off += 1024 * 128;
  float* scale1 = ws + off; off += 128;
  float* shift1 = ws + off; off += 128;
  float* scale2 = ws + off; off += 128;
  float* shift2 = ws + off; off += 128;
  int* fps_idx = (int*)(ws + off); off += (size_t)B_ * S_;
  int* knn_idx = (int*)(ws + off);

  const dim3 ggrid(N_ / 64, B_);
  gemm_kernel<CIN_,  true,  false><<<ggrid, 128, 0, stream>>>(
      feats, W1, b1, nullptr, nullptr, y1, psum1, psq1);
  finalize_kernel<<<1, 128, 0, stream>>>(psum1, psq1, 1024, g1, be1,
                                         scale1, shift1, (float)(B_ * N_));
  gemm_kernel<COUT_, false, true ><<<ggrid, 128, 0, stream>>>(
      y1, W2, b2, scale1, shift1, y2, psum2, psq2);
  finalize_kernel<<<1, 128, 0, stream>>>(psum2, psq2, 1024, g2, be2,
                                         scale2, shift2, (float)(B_ * N_));
  fps_kernel<<<B_, 1024, 0, stream>>>(points, fps_idx, cent);
  knn_kernel<<<(B_ * S_) / 8, 256, 0, stream>>>(points, cent, knn_idx);
  gather_kernel<<<dim3(S_, B_), 128, 0, stream>>>(y2, scale2, shift2, knn_idx, out);
}